// clDiceLoss_309237645902
// MI455X (gfx1250) — compile-verified
//
#include <hip/hip_runtime.h>
#include <math.h>

// ---------------------------------------------------------------------------
// Problem constants (match the JAX reference)
// ---------------------------------------------------------------------------
#define H_IMG 512
#define W_IMG 512
#define N_IMG 32
#define ITERS 10
#define SMOOTH 1e-5f

#define TILE 128
#define HALO (2 * ITERS)          // 20: each iteration consumes a 2-px ring
#define LT   (TILE + 2 * HALO)    // 168
#define RS   (LT + 1)             // 169 (odd row stride -> fewer bank conflicts)
#define NTHREADS 512              // 16 wave32s per block
#define TILES_PER_IMG 16          // (512/128)^2
#define NTILES (N_IMG * TILES_PER_IMG)  // 512 workgroups

// Dynamic LDS layout (floats):  S | E | P(inner pred skel)
#define OFF_S 0
#define OFF_E (RS * LT)                       // 28392
#define OFF_P (2 * RS * LT)                   // 56784
#define LDS_FLOATS (2 * RS * LT + TILE * TILE)  // 73168 floats = 292,672 B < 320 KB

// ---------------------------------------------------------------------------
// CDNA5 feature probes
// ---------------------------------------------------------------------------
#if defined(__AMDGCN__) && __has_builtin(__builtin_amdgcn_global_load_async_to_lds_b32)
#define HAVE_ASYNC 1
#else
#define HAVE_ASYNC 0
#endif

#if defined(__gfx1250__) && __has_builtin(__builtin_amdgcn_wmma_f32_16x16x4_f32)
#define HAVE_WMMA4 1
#else
#define HAVE_WMMA4 0
#endif

#if HAVE_ASYNC
typedef __attribute__((address_space(1))) int global_i32;  // printed as "__device__ int"
typedef __attribute__((address_space(3))) int lds_i32;     // printed as "__shared__ int"
#endif

typedef __attribute__((ext_vector_type(2))) float v2f;
typedef __attribute__((ext_vector_type(8))) float v8f;

__device__ __forceinline__ bool inimg(int gy, int gx) {
  return ((unsigned)gy < (unsigned)H_IMG) && ((unsigned)gx < (unsigned)W_IMG);
}

// Issue an async global->LDS copy for valid cells; sentinel-fill out-of-image
// cells.  Sentinel +2.0 acts as +inf for min-pool (all live data is in [0,1]).
__device__ __forceinline__ void tile_cell_load(const float* g, float* l, bool valid) {
  if (valid) {
#if HAVE_ASYNC
    __builtin_amdgcn_global_load_async_to_lds_b32((global_i32*)g, (lds_i32*)l, 0, 0);
#else
    *l = *g;
#endif
  } else {
    *l = 2.0f;
  }
}

__device__ __forceinline__ void wait_async_and_sync() {
#if HAVE_ASYNC
#if __has_builtin(__builtin_amdgcn_s_wait_asynccnt)
  __builtin_amdgcn_s_wait_asynccnt(0);
#else
  asm volatile("s_wait_asynccnt 0" ::: "memory");
#endif
#endif
  __syncthreads();
}

__device__ void load_tile(const float* __restrict__ src, float* lds, int base,
                          int oy, int ox) {
  for (int idx = threadIdx.x; idx < LT * LT; idx += NTHREADS) {
    int ly = idx / LT;
    int lx = idx - ly * LT;
    int gy = oy + ly, gx = ox + lx;
    bool v = inimg(gy, gx);
    const float* gp = v ? (src + gy * W_IMG + gx) : src;
    tile_cell_load(gp, &lds[base + ly * RS + lx], v);
  }
}

// 10 fused skeletonization iterations entirely in LDS.
// Valid region shrinks by 1 per pass (2 per iteration); HALO = 20 exactly
// covers it so the inner 128x128 is exact at the end.
// SAME padding: S holds +2.0 at image-OOB cells (ignored by min), E is
// rewritten to -2.0 at image-OOB cells (ignored by max).
__device__ void skeletonize(float* lds, int oy, int ox) {
  float* S = lds + OFF_S;
  float* E = lds + OFF_E;
  for (int it = 0; it < ITERS; ++it) {
    // erode: E = min3x3(S)
    int r = 2 * it + 1;
    int n = LT - 2 * r;
    for (int idx = threadIdx.x; idx < n * n; idx += NTHREADS) {
      int y = r + idx / n;
      int x = r + (idx - (idx / n) * n);
      const float* c = S + y * RS + x;
      float m = c[0];
      m = fminf(m, c[-1]);      m = fminf(m, c[1]);
      m = fminf(m, c[-RS - 1]); m = fminf(m, c[-RS]); m = fminf(m, c[-RS + 1]);
      m = fminf(m, c[RS - 1]);  m = fminf(m, c[RS]);  m = fminf(m, c[RS + 1]);
      E[y * RS + x] = inimg(oy + y, ox + x) ? m : -2.0f;
    }
    __syncthreads();
    // dilate-of-eroded computed on the fly + skeleton update (in place on S)
    int r2 = r + 1;
    int n2 = LT - 2 * r2;
    for (int idx = threadIdx.x; idx < n2 * n2; idx += NTHREADS) {
      int y = r2 + idx / n2;
      int x = r2 + (idx - (idx / n2) * n2);
      const float* c = E + y * RS + x;
      float t = c[0];
      t = fmaxf(t, c[-1]);      t = fmaxf(t, c[1]);
      t = fmaxf(t, c[-RS - 1]); t = fmaxf(t, c[-RS]); t = fmaxf(t, c[-RS + 1]);
      t = fmaxf(t, c[RS - 1]);  t = fmaxf(t, c[RS]);  t = fmaxf(t, c[RS + 1]);
      float s = S[y * RS + x];
      float e = c[0];
      float v = s - (s - e) * t;
      v = fminf(fmaxf(v, 0.0f), 1.0f);
      S[y * RS + x] = inimg(oy + y, ox + x) ? v : 2.0f;
    }
    __syncthreads();
  }
}

// ---------------------------------------------------------------------------
// Kernel 1: one block per 128x128 tile. Produces 6 partial sums per tile:
//   [inter, sum_pred, sum_tgt, skel_inter, sum_pred_skel, sum_tgt_skel]
// ---------------------------------------------------------------------------
__global__ __launch_bounds__(NTHREADS) void cldice_tile_kernel(
    const float* __restrict__ logits, const float* __restrict__ target,
    float* __restrict__ tile_out) {
  extern __shared__ float lds[];
  const int bid = blockIdx.x;        // 0..511
  const int img = bid >> 4;
  const int t = bid & 15;
  const int ty = t >> 2, tx = t & 3;
  const int oy = ty * TILE - HALO, ox = tx * TILE - HALO;
  const float* L = logits + (size_t)img * (H_IMG * W_IMG);
  const float* T = target + (size_t)img * (H_IMG * W_IMG);

  float a_inter = 0.f, a_p = 0.f, a_t = 0.f;
  float a_si = 0.f, a_sp = 0.f, a_st = 0.f;

  // -------- phase 1: pred (sigmoid) in S, target in E --------
  load_tile(L, lds, OFF_S, oy, ox);
  load_tile(T, lds, OFF_E, oy, ox);
  wait_async_and_sync();

  for (int idx = threadIdx.x; idx < LT * LT; idx += NTHREADS) {
    int ly = idx / LT;
    int lx = idx - ly * LT;
    if (inimg(oy + ly, ox + lx)) {
      float x = lds[OFF_S + ly * RS + lx];
      lds[OFF_S + ly * RS + lx] = 1.0f / (1.0f + __expf(-x));
    }
  }
  __syncthreads();

  // dice partials over the inner 128x128 (always fully in-image)
  for (int idx = threadIdx.x; idx < TILE * TILE; idx += NTHREADS) {
    int iy = idx >> 7, ix = idx & (TILE - 1);
    float p = lds[OFF_S + (HALO + iy) * RS + HALO + ix];
    float tg = lds[OFF_E + (HALO + iy) * RS + HALO + ix];
    a_p += p;
    a_t += tg;
    a_inter += p * tg;
  }
  __syncthreads();  // protect E before erode overwrites it

  skeletonize(lds, oy, ox);  // pred skeleton now in S (inner region exact)

  for (int idx = threadIdx.x; idx < TILE * TILE; idx += NTHREADS) {
    int iy = idx >> 7, ix = idx & (TILE - 1);
    lds[OFF_P + idx] = lds[OFF_S + (HALO + iy) * RS + HALO + ix];
  }
  __syncthreads();

  // -------- phase 2: target skeleton --------
  load_tile(T, lds, OFF_S, oy, ox);
  wait_async_and_sync();
  skeletonize(lds, oy, ox);

  for (int idx = threadIdx.x; idx < TILE * TILE; idx += NTHREADS) {
    int iy = idx >> 7, ix = idx & (TILE - 1);
    float ps = lds[OFF_P + idx];
    float ts = lds[OFF_S + (HALO + iy) * RS + HALO + ix];
    a_sp += ps;
    a_st += ts;
    a_si += ps * ts;
  }
  __syncthreads();

  // -------- deterministic block tree-reduction (reuse E space) --------
  float acc[6] = {a_inter, a_p, a_t, a_si, a_sp, a_st};
  float* red = lds + OFF_E;
  for (int q = 0; q < 6; ++q) red[q * NTHREADS + threadIdx.x] = acc[q];
  __syncthreads();
  for (int sft = NTHREADS >> 1; sft > 0; sft >>= 1) {
    if (threadIdx.x < sft) {
      for (int q = 0; q < 6; ++q)
        red[q * NTHREADS + threadIdx.x] += red[q * NTHREADS + threadIdx.x + sft];
    }
    __syncthreads();
  }
  if (threadIdx.x < 6) tile_out[bid * 6 + threadIdx.x] = red[threadIdx.x * NTHREADS];
}

// ---------------------------------------------------------------------------
// Kernel 2: single wave32. Per-image 16-tile sums done with
// V_WMMA_F32_16X16X4_F32 against an all-ones B matrix (D[i][*] = sum_k A[i][k]),
// chained over 4 K-groups; two chains cover images 0-15 and 16-31.
// ---------------------------------------------------------------------------
__global__ __launch_bounds__(32) void cldice_final_kernel(
    const float* __restrict__ ts, float* __restrict__ out) {
  const int lane = threadIdx.x;
  float dtot = 0.f, stot = 0.f;
#if HAVE_WMMA4
  for (int c = 0; c < 2; ++c) {
    v8f acc[6] = {};
    const int row = c * 16 + (lane & 15);        // A-matrix M = image
    const int half = (lane >> 4) << 1;           // lanes 16-31 hold K=2,3
    v2f ones;
    ones.x = 1.0f;
    ones.y = 1.0f;
    for (int g = 0; g < 4; ++g) {                // K-groups of 4 tiles
      const int t0 = 4 * g + half;
      for (int q = 0; q < 6; ++q) {
        v2f a;                                   // A 16x4 f32: V0=K0, V1=K1 / V0=K2, V1=K3
        a.x = ts[(row * TILES_PER_IMG + t0) * 6 + q];
        a.y = ts[(row * TILES_PER_IMG + t0 + 1) * 6 + q];
        acc[q] = __builtin_amdgcn_wmma_f32_16x16x4_f32(
            false, a, false, ones, (short)0, acc[q], false, false);
      }
    }
    // C/D layout: VGPR r, lanes 0-15 -> M=r ; lanes 16-31 -> M=8+r.
    float d = 0.f, s = 0.f;
    for (int r = 0; r < 8; ++r) {
      float inter = acc[0][r], p = acc[1][r], tg = acc[2][r];
      float si = acc[3][r], sp = acc[4][r], st = acc[5][r];
      d += (2.0f * inter + SMOOTH) / (p + tg + SMOOTH);
      s += (2.0f * si + SMOOTH) / (sp + st + SMOOTH);
    }
    dtot += __shfl(d, 0) + __shfl(d, 16);   // rows 0-7 + rows 8-15 of this chain
    stot += __shfl(s, 0) + __shfl(s, 16);
  }
#else
  float acc6[6] = {0, 0, 0, 0, 0, 0};
  for (int t = 0; t < TILES_PER_IMG; ++t)
    for (int q = 0; q < 6; ++q) acc6[q] += ts[(lane * TILES_PER_IMG + t) * 6 + q];
  float d = (2.0f * acc6[0] + SMOOTH) / (acc6[1] + acc6[2] + SMOOTH);
  float s = (2.0f * acc6[3] + SMOOTH) / (acc6[4] + acc6[5] + SMOOTH);
  for (int o = 16; o > 0; o >>= 1) {
    d += __shfl_down(d, o);
    s += __shfl_down(s, o);
  }
  dtot = __shfl(d, 0);
  stot = __shfl(s, 0);
#endif
  const float dice = dtot * (1.0f / 32.0f);
  const float sdice = stot * (1.0f / 32.0f);
  const float loss = 0.5f * (1.0f - dice) + 0.5f * (1.0f - sdice);
  if (lane == 0) {
    out[0] = loss;
    out[1] = dice;
    out[2] = sdice;
  }
}

// ---------------------------------------------------------------------------
extern "C" void kernel_launch(void* const* d_in, const int* in_sizes, int n_in,
                              void* d_out, int out_size, void* d_ws, size_t ws_size,
                              hipStream_t stream) {
  const float* logits = (const float*)d_in[0];
  const float* target = (const float*)d_in[1];
  float* out = (float*)d_out;
  float* tile_sums = (float*)d_ws;  // NTILES * 6 floats = 12 KB

  const size_t shbytes = (size_t)LDS_FLOATS * sizeof(float);  // 292,672 B
  cldice_tile_kernel<<<dim3(NTILES), dim3(NTHREADS), shbytes, stream>>>(
      logits, target, tile_sums);
  cldice_final_kernel<<<dim3(1), dim3(32), 0, stream>>>(tile_sums, out);
}